// GNNSAGE_9251359555754
// MI455X (gfx1250) — compile-verified
//
#include <hip/hip_runtime.h>

// ---------------------------------------------------------------------------
// 2-layer bipartite GraphSAGE for MI455X (gfx1250).
//
// Perf model: 8.2 GFLOP of GEMM (negligible) vs ~8 GB of L2-resident
// gather/atomic traffic for the 4 segment-mean passes -> aggregation bound.
// GEMMs run in full fp32 via V_WMMA_F32_16X16X4_F32 (matrix pipe without
// precision loss; lower precision buys nothing end-to-end here).
// Wave32 fragment layouts per CDNA5 ISA 7.12.2.
// ---------------------------------------------------------------------------

typedef __attribute__((ext_vector_type(2))) float v2f;
typedef __attribute__((ext_vector_type(8))) float v8f;

#define NU_  200000
#define NM_  50000
#define EDG_ 4000000
#define DD   64

// ---------------------------------------------------------------- zero fill
__global__ void zero_f32(float* __restrict__ p, long long n) {
  long long i = (long long)blockIdx.x * blockDim.x + threadIdx.x;
  if (i < n) p[i] = 0.0f;
}

// ------------------------------------------------- edge scatter (segment sum)
// thread = (edge, feature). A wave32 covers 32 consecutive features of ONE
// edge: the gather xsrc[src[e]*64 + f] and the atomic target are each a
// single 128B cacheline per wave. cnt may be null (layer 2 reuses degrees).
__global__ __launch_bounds__(256) void scatter_accum(
    const float* __restrict__ xsrc, const int* __restrict__ src,
    const int* __restrict__ dst, float* __restrict__ agg,
    float* __restrict__ cnt, int nedges) {
  long long tid = (long long)blockIdx.x * blockDim.x + threadIdx.x;
  long long e = tid >> 6;
  int f = (int)(tid & 63);
  if (e >= nedges) return;
  int s = src[e];
  int d = dst[e];
  float v = xsrc[(long long)s * DD + f];
  atomicAdd(&agg[(long long)d * DD + f], v);
  if (cnt != nullptr && f == 0) atomicAdd(&cnt[d], 1.0f);
}

// ------------------------------------------------------------- fused SAGE GEMM
// out[r,:] = lrelu( (agg[r,:]/max(cnt[r],1)) @ Wl + b + xdst[r,:] @ Wr )
//
// One wave -> 16x64 output tile. Single K=128 GEMM: A = [mean | xdst] (16x128),
// B = [Wl ; Wr] (128x64), 32 steps of V_WMMA_F32_16X16X4_F32 x 4 N-tiles.
//
// Weight panel is PAIR-INTERLEAVED in LDS:
//   sW[(K>>1)*SWP + 2*c + (K&1)] = B[K][c]
// so a B fragment {B[k][c], B[k+1][c]} is one aligned ds_load_b64 straight
// into an even VGPR pair. SWP = 160 dwords (== 32 mod 64): lanes 0-15
// (K-pair row p) hit banks {2c}, lanes 16-31 (row p+1, +160 dw) hit banks
// {2c+32}; the 16-wide c windows keep the sets disjoint -> conflict-free.
//
// All B addresses are expressed as CONSTANT offsets from one per-lane base
// (&sW[half*SWP + 2*lpos]); the 40KB panel fits the 16-bit ds immediate
// offset, so the unrolled loop needs zero per-step VALU address math.
#define SWP 160

__global__ __launch_bounds__(256) void sage_gemm(
    const float* __restrict__ agg, const float* __restrict__ cnt,
    const float* __restrict__ xdst, const float* __restrict__ Wl,
    const float* __restrict__ Wr, const float* __restrict__ bias,
    float* __restrict__ out, int n) {
  __shared__ float sW[64 * SWP];   // 64 K-pair rows (K = 0..127), 40KB
  __shared__ float sB[64];

  const int t = threadIdx.x;
  // stage Wl (K = 0..63) and Wr (K = 64..127), pair-interleaved
  for (int idx = t; idx < 64 * 64; idx += 256) {
    int k = idx >> 6, j = idx & 63;
    sW[(k >> 1) * SWP + 2 * j + (k & 1)]        = Wl[idx];      // K = k
    sW[(32 + (k >> 1)) * SWP + 2 * j + (k & 1)] = Wr[idx];      // K = 64 + k
  }
  if (t < 64) sB[t] = bias[t];
  __syncthreads();

  const int wave = t >> 5;          // 8 waves, 128 rows per block
  const int lane = t & 31;
  const int half = lane >> 4;       // lane-half per ISA A/B/C layouts
  const int lpos = lane & 15;

  const long long rowbase = (long long)blockIdx.x * 128 + wave * 16;
  long long row = rowbase + lpos;
  long long rc  = row < n ? row : (long long)(n - 1);  // clamp for safe loads

  const float inv = 1.0f / fmaxf(cnt[rc], 1.0f);
  // per-lane bases; every load below is base + compile-time-constant offset
  const float* aBase = agg  + rc * DD + 2 * half;   // A: K = 4s(+1) + 2*half
  const float* xBase = xdst + rc * DD + 2 * half;
  const float* bBase = &sW[half * SWP + 2 * lpos];  // B: row 2s+half, col lpos

  v8f acc[4];
  acc[0] = v8f{}; acc[1] = v8f{}; acc[2] = v8f{}; acc[3] = v8f{};

  // ---- K = 0..63 : mean part (A = agg * inv) --------------------------------
  #pragma unroll
  for (int s = 0; s < 16; ++s) {
    v2f a = *(const v2f*)(aBase + 4 * s);     // 8B aligned
    a.x *= inv; a.y *= inv;
    #pragma unroll
    for (int nt = 0; nt < 4; ++nt) {
      v2f b = *(const v2f*)(bBase + 2 * s * SWP + 32 * nt);  // imm-offset ds
      acc[nt] = __builtin_amdgcn_wmma_f32_16x16x4_f32(
          false, a, false, b, (short)0, acc[nt], false, false);
    }
  }

  // ---- K = 64..127 : root part (A = xdst) -----------------------------------
  #pragma unroll
  for (int s = 16; s < 32; ++s) {
    v2f a = *(const v2f*)(xBase + 4 * (s - 16));
    #pragma unroll
    for (int nt = 0; nt < 4; ++nt) {
      v2f b = *(const v2f*)(bBase + 2 * s * SWP + 32 * nt);
      acc[nt] = __builtin_amdgcn_wmma_f32_16x16x4_f32(
          false, a, false, b, (short)0, acc[nt], false, false);
    }
  }

  // ---- epilogue: bias + leaky_relu(0.01) + store ----------------------------
  // C/D layout: VGPR j holds D[M = j + 8*half][N = lpos (+16*nt)]
  #pragma unroll
  for (int nt = 0; nt < 4; ++nt) {
    #pragma unroll
    for (int j = 0; j < 8; ++j) {
      long long r = rowbase + j + 8 * half;
      int c = 16 * nt + lpos;
      if (r < n) {
        float v = acc[nt][j] + sB[c];
        v = v > 0.0f ? v : 0.01f * v;
        out[r * DD + c] = v;
      }
    }
  }
}

// ---------------------------------------------------------------------------
extern "C" void kernel_launch(void* const* d_in, const int* in_sizes, int n_in,
                              void* d_out, int out_size, void* d_ws,
                              size_t ws_size, hipStream_t stream) {
  const float* x_user   = (const float*)d_in[0];
  const float* x_movie  = (const float*)d_in[1];
  const int*   e_user   = (const int*)d_in[2];
  const int*   e_movie  = (const int*)d_in[3];
  const float* W_l1_um  = (const float*)d_in[4];
  const float* b_l1_um  = (const float*)d_in[5];
  const float* W_r1_um  = (const float*)d_in[6];
  const float* W_l1_mu  = (const float*)d_in[7];
  const float* b_l1_mu  = (const float*)d_in[8];
  const float* W_r1_mu  = (const float*)d_in[9];
  const float* W_l2     = (const float*)d_in[10];
  const float* b_l2     = (const float*)d_in[11];
  const float* W_r2     = (const float*)d_in[12];

  // workspace layout (floats): agg_m | agg_u | cnt_m | cnt_u | h_m | h_u
  float* ws    = (float*)d_ws;
  float* agg_m = ws;                                  // NM*64
  float* agg_u = agg_m + (long long)NM_ * DD;         // NU*64
  float* cnt_m = agg_u + (long long)NU_ * DD;         // NM
  float* cnt_u = cnt_m + NM_;                         // NU
  float* h_m   = cnt_u + NU_;                         // NM*64
  float* h_u   = h_m + (long long)NM_ * DD;           // NU*64

  float* out_u2 = (float*)d_out;                      // NU*64 (first output)
  float* out_m2 = out_u2 + (long long)NU_ * DD;       // NM*64 (second output)

  const long long sthreads = (long long)EDG_ * 64;
  const int sblocks = (int)((sthreads + 255) / 256);

  // ---- layer 1 ----
  long long z1 = (long long)(NM_ + NU_) * DD + NM_ + NU_;  // agg_* + cnt_*
  zero_f32<<<(int)((z1 + 255) / 256), 256, 0, stream>>>(agg_m, z1);

  scatter_accum<<<sblocks, 256, 0, stream>>>(x_user, e_user, e_movie,
                                             agg_m, cnt_m, EDG_);
  scatter_accum<<<sblocks, 256, 0, stream>>>(x_movie, e_movie, e_user,
                                             agg_u, cnt_u, EDG_);

  sage_gemm<<<(NM_ + 127) / 128, 256, 0, stream>>>(
      agg_m, cnt_m, x_movie, W_l1_um, W_r1_um, b_l1_um, h_m, NM_);
  sage_gemm<<<(NU_ + 127) / 128, 256, 0, stream>>>(
      agg_u, cnt_u, x_user, W_l1_mu, W_r1_mu, b_l1_mu, h_u, NU_);

  // ---- layer 2 (degrees unchanged: reuse cnt_*) ----
  long long z2 = (long long)(NM_ + NU_) * DD;              // agg_* only
  zero_f32<<<(int)((z2 + 255) / 256), 256, 0, stream>>>(agg_m, z2);

  scatter_accum<<<sblocks, 256, 0, stream>>>(h_u, e_user, e_movie,
                                             agg_m, nullptr, EDG_);
  scatter_accum<<<sblocks, 256, 0, stream>>>(h_m, e_movie, e_user,
                                             agg_u, nullptr, EDG_);

  sage_gemm<<<(NM_ + 127) / 128, 256, 0, stream>>>(
      agg_m, cnt_m, h_m, W_l2, W_r2, b_l2, out_m2, NM_);
  sage_gemm<<<(NU_ + 127) / 128, 256, 0, stream>>>(
      agg_u, cnt_u, h_u, W_l2, W_r2, b_l2, out_u2, NU_);
}